// Encoder_58669253263706
// MI455X (gfx1250) — compile-verified
//
#include <hip/hip_runtime.h>
#include <math.h>

// Neural-CDE RK4 scan, persistent per-workgroup time march, fp32 WMMA (gfx1250).
// B=256, T=32, IN=16, H=SH=128, OUT=64, NL=3.
// Small-layer weights (W0 + 3x Wl = 256KB) live in CDNA5's 320KB LDS with
// conflict-free padded stride; Wf (1MB) streams from L2.

namespace {
constexpr int B_T = 256, T_T = 32, IN = 16, H = 128, SH = 128, OUTW = 64, TM1 = 31;
constexpr int MT = 16;          // batch rows per workgroup (WMMA M)
constexpr int THREADS = 256;    // 8 waves of 32
constexpr int SA = 132;         // padded LDS row stride (floats): bank = (4n+k)&63

typedef __attribute__((ext_vector_type(2))) float v2f;
typedef __attribute__((ext_vector_type(8))) float v8f;

__device__ __forceinline__ v8f wmma4(v2f a, v2f b, v8f c) {
  // D = A(16x4) * B(4x16) + C  (fp32)
  return __builtin_amdgcn_wmma_f32_16x16x4_f32(false, a, false, b, (short)0, c,
                                               false, false);
}

// Accumulate D(16x16) = A(16xK, row-major lda) * W^T where W is (N x K)
// row-major with row stride ldw; 'n' is this lane's output column.
__device__ __forceinline__ v8f gemm_tile(const float* A, int lda, const float* W,
                                         int ldw, int n, int K, int lane) {
  const int m = lane & 15;
  const int kh = (lane >> 4) << 1;  // lanes 0-15: k+{0,1} ; lanes 16-31: k+{2,3}
  v8f acc = {};
  #pragma unroll 8
  for (int k0 = 0; k0 < K; k0 += 4) {
    v2f a = *(const v2f*)(A + m * lda + k0 + kh);
    v2f b = *(const v2f*)(W + (size_t)n * ldw + k0 + kh);
    acc = wmma4(a, b, acc);
  }
  return acc;
}

// dst = relu(src @ Wlds^T + bias); src/Wlds in LDS (stride SA); one tile/wave.
__device__ __forceinline__ void mlp_layer(const float* src, const float* Wlds,
                                          const float* bias, float* dst,
                                          int wid, int lane) {
  const int mD = lane & 15, half = lane >> 4;
  const int n = wid * 16 + mD;
  v8f acc = gemm_tile(src, SA, Wlds, SA, n, SH, lane);
  const float bn = bias[n];
  #pragma unroll
  for (int v = 0; v < 8; ++v) {
    float x = acc[v] + bn;
    dst[(v + 8 * half) * SA + n] = fmaxf(x, 0.f);
  }
}

// out[b, t, :] = sigmoid(z @ W_last^T + b_last); OUT=64 -> waves 0..3
__device__ __forceinline__ void out_proj(const float* zb, const float* W_last,
                                         const float* b_last, float* out,
                                         int b0r, int t, int wid, int lane) {
  if (wid < 4) {
    const int mD = lane & 15, half = lane >> 4;
    const int n = wid * 16 + mD;
    v8f acc = gemm_tile(zb, SA, W_last, H, n, H, lane);
    const float bn = b_last[n];
    #pragma unroll
    for (int v = 0; v < 8; ++v) {
      int m = v + 8 * half;
      float x = acc[v] + bn;
      out[((size_t)(b0r + m) * T_T + t) * OUTW + n] = 1.f / (1.f + expf(-x));
    }
  }
}
}  // namespace

__global__ __launch_bounds__(THREADS) void ncde_wmma_kernel(
    const float* __restrict__ coeffs,                                  // (B,31,64)
    const float* __restrict__ W_init, const float* __restrict__ b_init,// (128,16),(128)
    const float* __restrict__ W0, const float* __restrict__ b0,        // (128,128),(128)
    const float* __restrict__ Wl, const float* __restrict__ bl,        // (3,128,128),(3,128)
    const float* __restrict__ Wf, const float* __restrict__ bf,        // (2048,128),(2048)
    const float* __restrict__ W_last, const float* __restrict__ b_last,// (64,128),(64)
    float* __restrict__ out) {                                         // (B,32,64)
  // LDS-resident weights for the 4 latency-critical hidden layers.
  __shared__ __align__(16) float wlds[4 * H * SA];   // 270336 B
  __shared__ __align__(16) float zbuf[MT * SA];      // state z (padded, WMMA A)
  __shared__ __align__(16) float kbuf[3][MT * H];    // k1,k2,k3 (dense)
  __shared__ __align__(16) float actI[MT * SA];      // ping (WMMA A)
  __shared__ __align__(16) float actA[MT * SA];      // pong (WMMA A)
  __shared__ __align__(16) float dxs[MT * IN];       // dx (also x0 staging)

  const int tid = threadIdx.x;
  const int lane = tid & 31;
  const int wid = tid >> 5;
  const int b0r = blockIdx.x * MT;
  const int mD = lane & 15;
  const int half = lane >> 4;
  constexpr float third = 1.f / 3.f;

  // ---- one-time weight preload into LDS (padded stride, conflict-free) ----
  for (int e = tid; e < 4 * H * SH; e += THREADS) {
    int mat = e >> 14;           // 0..3
    int r = (e >> 7) & (H - 1);  // output row n
    int k = e & (SH - 1);
    const float* src = (mat == 0) ? W0 : (Wl + (size_t)(mat - 1) * SH * SH);
    wlds[(size_t)mat * H * SA + r * SA + k] = src[(size_t)r * SH + k];
  }

  // ---- z0 = x0 @ W_init^T + b_init ; x0 = a[:,0,:] = coeffs[b][0][0:16] ----
  {
    int m = tid >> 4, i = tid & 15;
    dxs[tid] = coeffs[((size_t)(b0r + m) * TM1) * (4 * IN) + i];
  }
  __syncthreads();
  {
    const int n = wid * 16 + mD;
    v8f acc = gemm_tile(dxs, IN, W_init, IN, n, IN, lane);
    const float bn = b_init[n];
    #pragma unroll
    for (int v = 0; v < 8; ++v)
      zbuf[(v + 8 * half) * SA + n] = acc[v] + bn;
  }
  __syncthreads();
  out_proj(zbuf, W_last, b_last, out, b0r, 0, wid, lane);
  __syncthreads();

  for (int step = 0; step < TM1; ++step) {
    #pragma unroll 1
    for (int s = 0; s < 4; ++s) {
      const float frac = (s == 0) ? 0.f
                       : (s == 1) ? third
                       : (s == 2) ? (2.f * third) : 1.f;
      // dx = b + (c + d*frac)*frac  (one element per thread, 16x16)
      {
        int m = tid >> 4, i = tid & 15;
        const float* cb = coeffs + ((size_t)(b0r + m) * TM1 + step) * (4 * IN);
        float bi = cb[IN + i], ci = cb[2 * IN + i], di = cb[3 * IN + i];
        dxs[tid] = bi + (ci + di * frac) * frac;
      }
      // zz for this RK stage: k1=kbuf[0], k2=kbuf[1], k3=kbuf[2]
      for (int e = tid; e < MT * H; e += THREADS) {
        int m = e >> 7, hc = e & (H - 1);
        float v = zbuf[m * SA + hc];
        if (s == 1)      v += kbuf[0][e] * third;
        else if (s == 2) v += kbuf[1][e] - kbuf[0][e] * third;
        else if (s == 3) v += kbuf[0][e] - kbuf[1][e] + kbuf[2][e];
        actI[m * SA + hc] = v;
      }
      __syncthreads();

      // 4 hidden layers (relu), weights fully in LDS; one 16-col tile per wave
      mlp_layer(actI, wlds + 0 * H * SA, b0,          actA, wid, lane); __syncthreads();
      mlp_layer(actA, wlds + 1 * H * SA, bl + 0 * SH, actI, wid, lane); __syncthreads();
      mlp_layer(actI, wlds + 2 * H * SA, bl + 1 * SH, actA, wid, lane); __syncthreads();
      mlp_layer(actA, wlds + 3 * H * SA, bl + 2 * SH, actI, wid, lane); __syncthreads();

      // f = tanh(h @ Wf^T + bf) fused with einsum('bhi,bi->bh', f, dx).
      // Stages 0..2 write k_{s+1}; stage 3 fuses k4 + RK combine into z.
      {
        float dxr[8];
        #pragma unroll
        for (int v = 0; v < 8; ++v) dxr[v] = dxs[(v + 8 * half) * IN + mD];
        const int kh = half << 1;
        for (int t0 = wid; t0 < H; t0 += 16) {        // 2 tiles/iter for ILP
          const int t1 = t0 + 8;
          const int n0 = t0 * 16 + mD, n1 = t1 * 16 + mD;
          v8f acc0 = {}, acc1 = {};
          #pragma unroll 8
          for (int k0 = 0; k0 < SH; k0 += 4) {
            v2f a  = *(const v2f*)(actI + mD * SA + k0 + kh);
            v2f w0 = *(const v2f*)(Wf + (size_t)n0 * SH + k0 + kh);
            v2f w1 = *(const v2f*)(Wf + (size_t)n1 * SH + k0 + kh);
            acc0 = wmma4(a, w0, acc0);
            acc1 = wmma4(a, w1, acc1);
          }
          const float bf0 = bf[n0], bf1 = bf[n1];
          #pragma unroll
          for (int v = 0; v < 8; ++v) {
            const int m = v + 8 * half;
            float f0 = tanhf(acc0[v] + bf0) * dxr[v];
            float f1 = tanhf(acc1[v] + bf1) * dxr[v];
            // butterfly sum over i (16 lanes within each half of the wave)
            f0 += __shfl_xor(f0, 1); f0 += __shfl_xor(f0, 2);
            f0 += __shfl_xor(f0, 4); f0 += __shfl_xor(f0, 8);
            f1 += __shfl_xor(f1, 1); f1 += __shfl_xor(f1, 2);
            f1 += __shfl_xor(f1, 4); f1 += __shfl_xor(f1, 8);
            if (mD == 0) {
              if (s < 3) {
                kbuf[s][m * H + t0] = f0;
                kbuf[s][m * H + t1] = f1;
              } else {  // z += (k1 + 3*(k2+k3) + k4) / 8
                zbuf[m * SA + t0] +=
                    (kbuf[0][m * H + t0] +
                     3.f * (kbuf[1][m * H + t0] + kbuf[2][m * H + t0]) + f0) * 0.125f;
                zbuf[m * SA + t1] +=
                    (kbuf[0][m * H + t1] +
                     3.f * (kbuf[1][m * H + t1] + kbuf[2][m * H + t1]) + f1) * 0.125f;
              }
            }
          }
        }
      }
      __syncthreads();
    }
    out_proj(zbuf, W_last, b_last, out, b0r, step + 1, wid, lane);
    __syncthreads();
  }
}

extern "C" void kernel_launch(void* const* d_in, const int* in_sizes, int n_in,
                              void* d_out, int out_size, void* d_ws, size_t ws_size,
                              hipStream_t stream) {
  (void)in_sizes; (void)n_in; (void)out_size; (void)d_ws; (void)ws_size;
  // setup_inputs order: times, coeffs, W_init, b_init, W0, b0, Wl, bl, Wf, bf, W_last, b_last
  const float* coeffs = (const float*)d_in[1];
  const float* W_init = (const float*)d_in[2];
  const float* b_init = (const float*)d_in[3];
  const float* W0     = (const float*)d_in[4];
  const float* b0     = (const float*)d_in[5];
  const float* Wl     = (const float*)d_in[6];
  const float* bl     = (const float*)d_in[7];
  const float* Wf     = (const float*)d_in[8];
  const float* bf     = (const float*)d_in[9];
  const float* W_last = (const float*)d_in[10];
  const float* b_last = (const float*)d_in[11];
  float* out = (float*)d_out;

  ncde_wmma_kernel<<<B_T / MT, THREADS, 0, stream>>>(
      coeffs, W_init, b_init, W0, b0, Wl, bl, Wf, bf, W_last, b_last, out);
}